// MeshPooling_34479997452437
// MI455X (gfx1250) — compile-verified
//
#include <hip/hip_runtime.h>
#include <stdint.h>

namespace {

typedef float v2f __attribute__((ext_vector_type(2)));
typedef float v8f __attribute__((ext_vector_type(8)));

constexpr int  kN  = 500000;
constexpr int  kE  = 3000000;
constexpr int  kC  = 128;
constexpr int  kM  = 125000;
constexpr int  kNP = 5;

// d_out layout (float elements), reference tuple order flattened:
constexpr long long OFF_NEWX   = 0;          // M*128 = 16,000,000
constexpr long long OFF_NEWE   = 16000000;   // 2E    =  6,000,000
constexpr long long OFF_NEWPOS = 22000000;   // M*2   =    250,000
constexpr long long OFF_NEWIDX = 22250000;   // M*5   =    625,000
constexpr long long OFF_NEWB   = 22875000;   // M     =    125,000
constexpr long long OFF_EDGEP  = 23000000;   // 2E    =  6,000,000
constexpr long long OFF_CL     = 29000000;   // N     =    500,000
constexpr long long OFF_BATCH  = 29500000;   // N     =    500,000

constexpr int CHUNK = 1024;
constexpr int NB    = (kE + CHUNK - 1) / CHUNK;   // 2930 sort/unique blocks

__device__ inline uint32_t lds_off(const void* p) {
  // LDS flat aperture has zero low 32 bits -> low bits are the DS offset.
  return (uint32_t)(uintptr_t)p;
}

// ---------------------------------------------------------------- zero accum
__global__ void k_zero(float* out, float* counts) {
  long long i = (long long)blockIdx.x * blockDim.x + threadIdx.x;
  long long stride = (long long)gridDim.x * blockDim.x;
  for (; i < 17125000LL; i += stride) {
    if (i < 16000000LL)       out[OFF_NEWX + i] = 0.0f;                  // x sums
    else if (i < 17000000LL)  out[OFF_NEWPOS + (i - 16000000LL)] = 0.0f; // pos/idx/batch sums
    else                      counts[i - 17000000LL] = 0.0f;
  }
}

// ------------------------------------------------- cluster + passthroughs
__global__ void k_cluster(const float* __restrict__ ia, const float* __restrict__ batch,
                          const int* __restrict__ ppool, int* __restrict__ clusterWs,
                          float* __restrict__ out) {
  int pidx = ppool[0];
  int i = blockIdx.x * blockDim.x + threadIdx.x;
  int stride = gridDim.x * blockDim.x;
  for (; i < kN; i += stride) {
    int c = (int)ia[(long long)i * kNP + pidx];
    clusterWs[i]        = c;
    out[OFF_CL + i]     = (float)c;
    out[OFF_BATCH + i]  = batch[i];
  }
}

// ------------------------- segment-sum scatter: async-LDS staged, f32 atomics
__global__ __launch_bounds__(256) void k_scatter(
    const float* __restrict__ x, const float* __restrict__ xpos,
    const float* __restrict__ ia, const float* __restrict__ batch,
    const int* __restrict__ clusterWs, float* __restrict__ out,
    float* __restrict__ counts) {
  __shared__ __align__(16) float sx[2][8][kC];   // double-buffered 8 rows x 128 f32
  const int wave = threadIdx.x >> 5;
  const int lane = threadIdx.x & 31;
  const int ntiles  = kN / 8;                    // 62500, exact
  const int tstride = gridDim.x;

  auto issue = [&](int tile, int buf) {
    int row = tile * 8 + wave;
    uint64_t g = (uint64_t)(uintptr_t)(x + (long long)row * kC + lane * 4);
    uint32_t l = lds_off(&sx[buf][wave][lane * 4]);
    // CDNA5 async copy global->LDS, 16B per lane, tracked by ASYNCcnt
    asm volatile("global_load_async_to_lds_b128 %0, %1, off"
                 :: "v"(l), "v"(g) : "memory");
  };

  int t = blockIdx.x;
  if (t < ntiles) issue(t, 0);
  int buf = 0;
  for (; t < ntiles; t += tstride, buf ^= 1) {
    int tn = t + tstride;
    if (tn < ntiles) {
      issue(tn, buf ^ 1);
      asm volatile("s_wait_asynccnt 1" ::: "memory");  // current buffer landed
    } else {
      asm volatile("s_wait_asynccnt 0" ::: "memory");
    }
    int row = t * 8 + wave;
    int c = __builtin_amdgcn_readfirstlane(clusterWs[row]);

    float4 v = *(const float4*)(&sx[buf][wave][lane * 4]);
    float* dst = out + OFF_NEWX + (long long)c * kC + lane * 4;
    atomicAdd(dst + 0, v.x); atomicAdd(dst + 1, v.y);
    atomicAdd(dst + 2, v.z); atomicAdd(dst + 3, v.w);

    if (lane < 2)       atomicAdd(out + OFF_NEWPOS + (long long)c * 2 + lane,
                                  xpos[(long long)row * 2 + lane]);
    else if (lane < 7)  atomicAdd(out + OFF_NEWIDX + (long long)c * 5 + (lane - 2),
                                  ia[(long long)row * kNP + (lane - 2)]);
    else if (lane == 7) atomicAdd(out + OFF_NEWB + c, batch[row]);
    else if (lane == 8) atomicAdd(counts + c, 1.0f);
  }
}

// ---------------------------------------------------------------- finalize
__global__ void k_finalx(float* out, const float* __restrict__ counts) {
  long long i = (long long)blockIdx.x * blockDim.x + threadIdx.x;
  long long stride = (long long)gridDim.x * blockDim.x;
  for (; i < (long long)kM * kC; i += stride) {
    int m = (int)(i >> 7);
    float inv = 1.0f / fmaxf(counts[m], 1.0f);
    out[OFF_NEWX + i] *= inv;
  }
}

__global__ void k_finale(float* out, const float* __restrict__ counts) {
  long long i = (long long)blockIdx.x * blockDim.x + threadIdx.x;
  long long stride = (long long)gridDim.x * blockDim.x;
  for (; i < (long long)kM * 8; i += stride) {
    int m = (int)(i >> 3), j = (int)(i & 7);
    float inv = 1.0f / fmaxf(counts[m], 1.0f);
    if (j < 2) {
      out[OFF_NEWPOS + (long long)m * 2 + j] *= inv;
    } else if (j < 7) {
      long long p = OFF_NEWIDX + (long long)m * 5 + (j - 2);
      out[p] = truncf(out[p] * inv);            // astype(int64) semantics
    } else {
      long long p = OFF_NEWB + m;
      out[p] = truncf(out[p] * inv);
    }
  }
}

// --------------------------------------------- edge keys + edge passthrough
__global__ void k_keys(const long long* __restrict__ eidx, const int* __restrict__ clusterWs,
                       unsigned long long* __restrict__ keysA, float* __restrict__ out) {
  long long g = (long long)blockIdx.x * blockDim.x + threadIdx.x;
  long long stride = (long long)gridDim.x * blockDim.x;
  for (; g < 2LL * kE; g += stride) {
    long long v = eidx[g];
    out[OFF_EDGEP + g] = (float)v;
    if (g < kE) {
      int s = (int)v;
      int d = (int)eidx[kE + g];
      keysA[g] = (unsigned long long)clusterWs[s] * (unsigned long long)kM +
                 (unsigned long long)clusterWs[d];
    }
  }
}

// ------------------------------------------------------- radix sort kernels
__global__ __launch_bounds__(256) void k_hist(const unsigned long long* __restrict__ keys,
                                              unsigned* __restrict__ blockHist, int shift) {
  __shared__ unsigned h[256];
  h[threadIdx.x] = 0;
  __syncthreads();
  long long base = (long long)blockIdx.x * CHUNK;
  for (int k = 0; k < 4; ++k) {
    long long e = base + k * 256 + threadIdx.x;
    if (e < kE) atomicAdd(&h[(unsigned)(keys[e] >> shift) & 255u], 1u);
  }
  __syncthreads();
  blockHist[(long long)blockIdx.x * 256 + threadIdx.x] = h[threadIdx.x];
}

// Per-digit block scan + 256-wide exclusive scan done as a tensor-core scan:
//   X = totals viewed 16x16;  escan = (Lstrict*X)*J + X*Ustrict
// All values are integers < 2^24, so f32 WMMA is exact.
__global__ __launch_bounds__(256) void k_scanhist(unsigned* blockHist, unsigned* digitBase) {
  int t = threadIdx.x;
  unsigned total = 0;
  for (int b = 0; b < NB; ++b) {
    unsigned v = blockHist[(long long)b * 256 + t];
    blockHist[(long long)b * 256 + t] = total;   // per-digit exclusive over blocks
    total += v;
  }
  __shared__ float s[256];
  __shared__ float ybuf[256];
  __shared__ float res[256];
  s[t] = (float)total;
  __syncthreads();

  if (t < 32) {                 // one fully-active wave: EXEC all-1s for WMMA
    const int m = t & 15;       // A-role: M row / B-role: N col
    const int h = t >> 4;       // lane half selects K pair {2h, 2h+1}
    // ---- Y = Lstrict * X   (K chained 4x V_WMMA_F32_16X16X4_F32)
    v8f acc = {};
#pragma unroll
    for (int k = 0; k < 4; ++k) {
      const int c0 = 4 * k + 2 * h;
      v2f a, b;
      a.x = (c0 + 0 < m) ? 1.0f : 0.0f;      // Lstrict(m, c0+0)
      a.y = (c0 + 1 < m) ? 1.0f : 0.0f;      // Lstrict(m, c0+1)
      b.x = s[(c0 + 0) * 16 + m];            // X(c0+0, n=m)
      b.y = s[(c0 + 1) * 16 + m];            // X(c0+1, n=m)
      acc = __builtin_amdgcn_wmma_f32_16x16x4_f32(false, a, false, b,
                                                  (short)0, acc, false, false);
    }
#pragma unroll
    for (int v = 0; v < 8; ++v) ybuf[(v + 8 * h) * 16 + m] = acc[v]; // D->LDS
    // ---- Z = X * Ustrict
    v8f acc2 = {};
#pragma unroll
    for (int k = 0; k < 4; ++k) {
      const int c0 = 4 * k + 2 * h;
      v2f a, b;
      a.x = s[m * 16 + c0 + 0];              // X(m, c0+0)
      a.y = s[m * 16 + c0 + 1];              // X(m, c0+1)
      b.x = (c0 + 0 < m) ? 1.0f : 0.0f;      // Ustrict(c0+0, n=m)
      b.y = (c0 + 1 < m) ? 1.0f : 0.0f;      // Ustrict(c0+1, n=m)
      acc2 = __builtin_amdgcn_wmma_f32_16x16x4_f32(false, a, false, b,
                                                   (short)0, acc2, false, false);
    }
    // ---- W = Y * J + Z   (J = all ones)
#pragma unroll
    for (int k = 0; k < 4; ++k) {
      const int c0 = 4 * k + 2 * h;
      v2f a, b;
      a.x = ybuf[m * 16 + c0 + 0];           // Y(m, c0+0)
      a.y = ybuf[m * 16 + c0 + 1];           // Y(m, c0+1)
      b.x = 1.0f; b.y = 1.0f;
      acc2 = __builtin_amdgcn_wmma_f32_16x16x4_f32(false, a, false, b,
                                                   (short)0, acc2, false, false);
    }
#pragma unroll
    for (int v = 0; v < 8; ++v) res[(v + 8 * h) * 16 + m] = acc2[v];
  }
  __syncthreads();
  digitBase[t] = (unsigned)res[t];           // exclusive digit base
}

// Stable scatter: single-wave blocks, intra-wave ranking via 32-bit ballots.
__global__ __launch_bounds__(32) void k_radix_scatter(
    const unsigned long long* __restrict__ in, unsigned long long* __restrict__ outk,
    const unsigned* __restrict__ blockHist, const unsigned* __restrict__ digitBase,
    int shift) {
  __shared__ unsigned base[256];
  __shared__ unsigned run[256];
  int lane = threadIdx.x;
  for (int d = lane; d < 256; d += 32) {
    base[d] = digitBase[d] + blockHist[(long long)blockIdx.x * 256 + d];
    run[d]  = 0;
  }
  __syncthreads();
  long long bstart = (long long)blockIdx.x * CHUNK;
  unsigned lt = (1u << lane) - 1u;
  for (int k = 0; k < CHUNK / 32; ++k) {
    long long e = bstart + (long long)k * 32 + lane;
    bool valid = e < kE;
    unsigned long long key = valid ? in[e] : 0ull;
    unsigned d = (unsigned)(key >> shift) & 255u;
    unsigned eq = (unsigned)__ballot(valid);
#pragma unroll
    for (int bit = 0; bit < 8; ++bit) {
      unsigned bal = (unsigned)__ballot((d >> bit) & 1u);
      eq &= ((d >> bit) & 1u) ? bal : ~bal;
    }
    unsigned rank = __popc(eq & lt);
    if (valid) {
      unsigned r0 = run[d];
      outk[base[d] + r0 + rank] = key;
      if ((eq & lt) == 0u) run[d] = r0 + __popc(eq);   // group leader advances
    }
    __syncthreads();
  }
}

// --------------------------------------------------------- unique (sorted)
__global__ __launch_bounds__(256) void k_ucount(const unsigned long long* __restrict__ keys,
                                                unsigned* __restrict__ bcnt) {
  __shared__ unsigned cnt;
  if (threadIdx.x == 0) cnt = 0;
  __syncthreads();
  long long base = (long long)blockIdx.x * CHUNK;
  unsigned local = 0;
  for (int k = 0; k < 4; ++k) {
    long long e = base + k * 256 + threadIdx.x;
    if (e < kE && (e == 0 || keys[e] != keys[e - 1])) local++;
  }
  atomicAdd(&cnt, local);
  __syncthreads();
  if (threadIdx.x == 0) bcnt[blockIdx.x] = cnt;
}

__global__ void k_uscan(const unsigned* __restrict__ bcnt, unsigned* __restrict__ bbase,
                        unsigned* __restrict__ utot) {
  unsigned runv = 0;
  for (int b = 0; b < NB; ++b) { bbase[b] = runv; runv += bcnt[b]; }
  utot[0] = runv;
}

__global__ __launch_bounds__(32) void k_uscatter(const unsigned long long* __restrict__ keys,
                                                 const unsigned* __restrict__ bbase,
                                                 unsigned long long* __restrict__ ukeys) {
  int lane = threadIdx.x;
  unsigned runv = bbase[blockIdx.x];
  long long bstart = (long long)blockIdx.x * CHUNK;
  unsigned lt = (1u << lane) - 1u;
  for (int k = 0; k < CHUNK / 32; ++k) {
    long long e = bstart + (long long)k * 32 + lane;
    bool flag = (e < kE) && (e == 0 || keys[e] != keys[e - 1]);
    unsigned bm = (unsigned)__ballot(flag);
    if (flag) ukeys[runv + __popc(bm & lt)] = keys[e];
    runv += __popc(bm);
  }
}

__global__ void k_decomp(const unsigned long long* __restrict__ ukeys,
                         const unsigned long long* __restrict__ sorted,
                         const unsigned* __restrict__ utot, float* __restrict__ out) {
  long long g = (long long)blockIdx.x * blockDim.x + threadIdx.x;
  long long stride = (long long)gridDim.x * blockDim.x;
  unsigned U = utot[0];
  unsigned long long kmax = sorted[kE - 1];      // == keys.max() after sort
  for (; g < kE; g += stride) {
    unsigned long long k = (g < (long long)U) ? ukeys[g] : kmax;
    out[OFF_NEWE + g]      = (float)(k / (unsigned long long)kM);
    out[OFF_NEWE + kE + g] = (float)(k % (unsigned long long)kM);
  }
}

} // namespace

extern "C" void kernel_launch(void* const* d_in, const int* in_sizes, int n_in,
                              void* d_out, int out_size, void* d_ws, size_t ws_size,
                              hipStream_t stream) {
  (void)in_sizes; (void)n_in; (void)out_size; (void)ws_size;
  const float*     x     = (const float*)d_in[0];
  const long long* eidx  = (const long long*)d_in[1];
  const float*     xpos  = (const float*)d_in[2];
  const float*     ia    = (const float*)d_in[3];
  const float*     batch = (const float*)d_in[4];
  const int*       ppool = (const int*)d_in[5];
  float*           out   = (float*)d_out;

  char* w = (char*)d_ws;
  size_t o = 0;
  auto alloc = [&](size_t bytes) -> void* {
    void* p = w + o;
    o = (o + bytes + 255) & ~(size_t)255;
    return p;
  };
  float*              counts    = (float*)alloc((size_t)kM * 4);
  int*                clusterWs = (int*)alloc((size_t)kN * 4);
  unsigned long long* keysA     = (unsigned long long*)alloc((size_t)kE * 8);
  unsigned long long* keysB     = (unsigned long long*)alloc((size_t)kE * 8);
  unsigned*           blockHist = (unsigned*)alloc((size_t)NB * 256 * 4);
  unsigned*           digitBase = (unsigned*)alloc(256 * 4);
  unsigned*           bcnt      = (unsigned*)alloc((size_t)NB * 4);
  unsigned*           bbase     = (unsigned*)alloc((size_t)NB * 4);
  unsigned*           utot      = (unsigned*)alloc(4);
  unsigned long long* ukeys     = (unsigned long long*)alloc((size_t)kE * 8);

  k_zero   <<<2048, 256, 0, stream>>>(out, counts);
  k_cluster<<<1954, 256, 0, stream>>>(ia, batch, ppool, clusterWs, out);
  k_scatter<<<1280, 256, 0, stream>>>(x, xpos, ia, batch, clusterWs, out, counts);
  k_finalx <<<2048, 256, 0, stream>>>(out, counts);
  k_finale <<<1024, 256, 0, stream>>>(out, counts);
  k_keys   <<<4096, 256, 0, stream>>>(eidx, clusterWs, keysA, out);

  unsigned long long* src = keysA;
  unsigned long long* dst = keysB;
  for (int p = 0; p < 5; ++p) {            // 40 bits covers keys < 2^34
    int shift = 8 * p;
    k_hist         <<<NB, 256, 0, stream>>>(src, blockHist, shift);
    k_scanhist     <<<1, 256, 0, stream>>>(blockHist, digitBase);
    k_radix_scatter<<<NB, 32, 0, stream>>>(src, dst, blockHist, digitBase, shift);
    unsigned long long* t = src; src = dst; dst = t;
  }
  // sorted keys now in `src`
  k_ucount  <<<NB, 256, 0, stream>>>(src, bcnt);
  k_uscan   <<<1, 1, 0, stream>>>(bcnt, bbase, utot);
  k_uscatter<<<NB, 32, 0, stream>>>(src, bbase, ukeys);
  k_decomp  <<<4096, 256, 0, stream>>>(ukeys, src, utot, out);
}